// SharedModule_82145544503553
// MI455X (gfx1250) — compile-verified
//
#include <hip/hip_runtime.h>

#define N_NODES  100000
#define N_EDGES  1600000
#define N_GRAPHS 512
#define DIN0     108
#define DH       98
#define DOUTF    120

typedef float v2f __attribute__((ext_vector_type(2)));
typedef float v8f __attribute__((ext_vector_type(8)));

// ---------------------------------------------------------------- utilities
__global__ void k_zero(int* p, int n) {
    int i = blockIdx.x * blockDim.x + threadIdx.x;
    int stride = gridDim.x * blockDim.x;
    for (; i < n; i += stride) p[i] = 0;
}

__global__ void k_hist(const int* __restrict__ dst, int* __restrict__ rowptr, int ne) {
    int e = blockIdx.x * 256 + threadIdx.x;
    if (e < ne) atomicAdd(&rowptr[dst[e] + 1], 1);
}

__global__ void k_scan1(int* __restrict__ data, int n, int* __restrict__ partials) {
    __shared__ int sh[256];
    int tid = threadIdx.x;
    int i = blockIdx.x * 256 + tid;
    sh[tid] = (i < n) ? data[i] : 0;
    __syncthreads();
    for (int off = 1; off < 256; off <<= 1) {
        int t = (tid >= off) ? sh[tid - off] : 0;
        __syncthreads();
        sh[tid] += t;
        __syncthreads();
    }
    if (i < n) data[i] = sh[tid];
    if (tid == 255) partials[blockIdx.x] = sh[255];
}

__global__ void k_scan2(int* partials, int nb) {
    if (threadIdx.x == 0 && blockIdx.x == 0) {
        int run = 0;
        for (int b = 0; b < nb; ++b) { int t = partials[b]; partials[b] = run; run += t; }
    }
}

__global__ void k_scan3(int* __restrict__ data, int n, const int* __restrict__ partials) {
    int i = blockIdx.x * 256 + threadIdx.x;
    if (i < n) data[i] += partials[blockIdx.x];
}

__global__ void k_fill(const int* __restrict__ src, const int* __restrict__ dst,
                       const int* __restrict__ rowptr, int* __restrict__ cursor,
                       int* __restrict__ colbuf, int ne) {
    int e = blockIdx.x * 256 + threadIdx.x;
    if (e < ne) {
        int d = dst[e];
        int p = atomicAdd(&cursor[d], 1);
        colbuf[rowptr[d] + p] = src[e];
    }
}

// ------------------------------------------------- fused GIN layer (WMMA f32)
// block: 256 threads = 8 waves; 64 nodes per block.
// Stage A: zero-padded weight panels -> LDS; CSR-gather aggregation -> LDS.
// Stage B: GEMM1 relu(h@w1+b1) -> LDS.  Stage C: GEMM2 (+b2, RReLU) -> global.
// All WMMA operands come from LDS: inner loops are branch-free.
template <int DIN>
__launch_bounds__(256)
__global__ void k_gin(const float* __restrict__ hin,
                      const float* __restrict__ w1, const float* __restrict__ b1,
                      const float* __restrict__ w2, const float* __restrict__ b2,
                      const int* __restrict__ rowptr, const int* __restrict__ cols,
                      float* __restrict__ hout, int nnodes) {
    __shared__ float hs[64 * 113];      // aggregated input, K padded to 112
    __shared__ float mid[64 * 113];     // relu output, K padded to 112
    __shared__ float w1s[112 * 112];    // zero-padded weight panels
    __shared__ float w2s[112 * 112];
    __shared__ float b1s[112];
    __shared__ float b2s[112];

    const int tid  = threadIdx.x;
    const int lane = tid & 31;
    const int wave = tid >> 5;
    const int lo   = lane & 15;
    const int hi   = lane >> 4;
    const int base = blockIdx.x * 64;
    const float slope = 0.5f * (0.125f + (1.0f / 3.0f));  // RReLU eval slope

    // ---- stage zero-padded weights + biases into LDS
    for (int idx = tid; idx < 112 * 112; idx += 256) {
        int r = idx / 112;
        int c = idx - r * 112;
        w1s[idx] = (r < DIN && c < DH) ? w1[r * DH + c] : 0.f;
        w2s[idx] = (r < DH  && c < DH) ? w2[r * DH + c] : 0.f;
    }
    if (tid < 112) {
        b1s[tid] = (tid < DH) ? b1[tid] : 0.f;
        b2s[tid] = (tid < DH) ? b2[tid] : 0.f;
    }

    // ---- aggregation gather into LDS (zero-padded to K=112)
    for (int i = wave * 8; i < wave * 8 + 8; ++i) {
        int node = base + i;
        float a0 = 0.f, a1 = 0.f, a2 = 0.f, a3 = 0.f;
        if (node < nnodes) {
            int e0 = rowptr[node], e1 = rowptr[node + 1];
            for (int e = e0; e < e1; ++e) {
                const float* xr = hin + (size_t)cols[e] * DIN;
                if (lane      < DIN) a0 += xr[lane];
                if (lane + 32 < DIN) a1 += xr[lane + 32];
                if (lane + 64 < DIN) a2 += xr[lane + 64];
                if (lane + 96 < DIN) a3 += xr[lane + 96];
            }
            const float* xs = hin + (size_t)node * DIN;
            if (lane      < DIN) a0 += xs[lane];
            if (lane + 32 < DIN) a1 += xs[lane + 32];
            if (lane + 64 < DIN) a2 += xs[lane + 64];
            if (lane + 96 < DIN) a3 += xs[lane + 96];
        }
        float* hr = hs + i * 113;
        hr[lane]      = (lane      < DIN) ? a0 : 0.f;
        hr[lane + 32] = (lane + 32 < DIN) ? a1 : 0.f;
        hr[lane + 64] = (lane + 64 < DIN) ? a2 : 0.f;
        if (lane + 96 < 112) hr[lane + 96] = (lane + 96 < DIN) ? a3 : 0.f;
    }
    __syncthreads();

    // ---- GEMM1: relu(hs @ w1 + b1) -> mid (cols >= 98 stored as zero)
    for (int t = wave; t < 28; t += 8) {          // 4 M-tiles x 7 N-tiles
        int tm = t / 7, tn = t % 7;
        int colB = tn * 16 + lo;
        v8f acc = {};
#pragma unroll
        for (int k0 = 0; k0 < 112; k0 += 4) {
            int ka = k0 + 2 * hi;
            v2f a, b;
            const float* ar = hs + (tm * 16 + lo) * 113 + ka;
            a.x = ar[0];
            a.y = ar[1];
            const float* br = w1s + ka * 112 + colB;
            b.x = br[0];
            b.y = br[112];
            acc = __builtin_amdgcn_wmma_f32_16x16x4_f32(false, a, false, b,
                                                        (short)0, acc, false, false);
        }
        float bias = b1s[colB];
#pragma unroll
        for (int r = 0; r < 8; ++r) {
            int row = tm * 16 + r + 8 * hi;
            float v = acc[r] + bias;
            v = v > 0.f ? v : 0.f;
            mid[row * 113 + tn * 16 + lo] = (colB < DH) ? v : 0.f;
        }
    }
    __syncthreads();

    // ---- GEMM2: (mid @ w2 + b2), RReLU -> global
    for (int t = wave; t < 28; t += 8) {
        int tm = t / 7, tn = t % 7;
        int colB = tn * 16 + lo;
        v8f acc = {};
#pragma unroll
        for (int k0 = 0; k0 < 112; k0 += 4) {
            int ka = k0 + 2 * hi;
            v2f a, b;
            const float* ar = mid + (tm * 16 + lo) * 113 + ka;
            a.x = ar[0];
            a.y = ar[1];
            const float* br = w2s + ka * 112 + colB;
            b.x = br[0];
            b.y = br[112];
            acc = __builtin_amdgcn_wmma_f32_16x16x4_f32(false, a, false, b,
                                                        (short)0, acc, false, false);
        }
        float bias = b2s[colB];
#pragma unroll
        for (int r = 0; r < 8; ++r) {
            int row  = tm * 16 + r + 8 * hi;
            int node = base + row;
            float v = acc[r] + bias;
            v = (v >= 0.f) ? v : v * slope;
            if (node < nnodes && colB < DH) hout[(size_t)node * DH + colB] = v;
        }
    }
}

// ------------------------------- final linear + sorted-batch segment-sum
__global__ void k_readout(const float* __restrict__ hin, const float* __restrict__ lw,
                          const float* __restrict__ lb, const int* __restrict__ batch,
                          float* __restrict__ g, int nnodes) {
    __shared__ float lws[DH * DOUTF];
    __shared__ float row[DH];
    const int tid = threadIdx.x;  // 128 threads
    for (int i = tid; i < DH * DOUTF; i += 128) lws[i] = lw[i];
    float bias = (tid < DOUTF) ? lb[tid] : 0.f;
    int base = blockIdx.x * 256;
    float acc = 0.f;
    int curg = -1;
    for (int nn = 0; nn < 256; ++nn) {
        int n = base + nn;
        if (n >= nnodes) break;
        __syncthreads();
        if (tid < DH) row[tid] = hin[(size_t)n * DH + tid];
        __syncthreads();
        int b = batch[n];
        if (b != curg) {  // uniform across block
            if (curg >= 0 && tid < DOUTF) atomicAdd(&g[curg * DOUTF + tid], acc);
            acc = 0.f;
            curg = b;
        }
        if (tid < DOUTF) {
            float v = bias;
            for (int k = 0; k < DH; ++k) v += row[k] * lws[k * DOUTF + tid];
            acc += v;
        }
    }
    if (curg >= 0 && tid < DOUTF) atomicAdd(&g[curg * DOUTF + tid], acc);
}

// ------------------------------- LayerNorm over 120 features per graph
__global__ void k_ln(const float* __restrict__ g, const float* __restrict__ lng,
                     const float* __restrict__ lnb, float* __restrict__ out) {
    __shared__ float red[128];
    int tid = threadIdx.x;
    int gi  = blockIdx.x;
    float v = (tid < DOUTF) ? g[gi * DOUTF + tid] : 0.f;
    red[tid] = v;
    __syncthreads();
    for (int s = 64; s > 0; s >>= 1) { if (tid < s) red[tid] += red[tid + s]; __syncthreads(); }
    float mu = red[0] / (float)DOUTF;
    __syncthreads();
    float d = (tid < DOUTF) ? (v - mu) : 0.f;
    red[tid] = d * d;
    __syncthreads();
    for (int s = 64; s > 0; s >>= 1) { if (tid < s) red[tid] += red[tid + s]; __syncthreads(); }
    float var = red[0] / (float)DOUTF;
    if (tid < DOUTF)
        out[gi * DOUTF + tid] = (v - mu) * rsqrtf(var + 1e-5f) * lng[tid] + lnb[tid];
}

// ---------------------------------------------------------------- launcher
extern "C" void kernel_launch(void* const* d_in, const int* in_sizes, int n_in,
                              void* d_out, int out_size, void* d_ws, size_t ws_size,
                              hipStream_t stream) {
    const float* x     = (const float*)d_in[0];
    const int*   ei    = (const int*)d_in[1];
    const int*   batch = (const int*)d_in[2];
    const float *gw1[5], *gb1[5], *gw2[5], *gb2[5];
    for (int l = 0; l < 5; ++l) {
        gw1[l] = (const float*)d_in[3 + l * 4 + 0];
        gb1[l] = (const float*)d_in[3 + l * 4 + 1];
        gw2[l] = (const float*)d_in[3 + l * 4 + 2];
        gb2[l] = (const float*)d_in[3 + l * 4 + 3];
    }
    const float* lw  = (const float*)d_in[23];
    const float* lb  = (const float*)d_in[24];
    const float* lng = (const float*)d_in[25];
    const float* lnb = (const float*)d_in[26];
    float* out = (float*)d_out;

    char* ws = (char*)d_ws;
    size_t off = 0;
    auto carve = [&](size_t bytes) -> char* {
        char* p = ws + off;
        off = (off + bytes + 255) & ~(size_t)255;
        return p;
    };
    int*   rowptr   = (int*)carve((N_NODES + 1) * sizeof(int));
    int*   cursor   = (int*)carve(N_NODES * sizeof(int));
    int*   colbuf   = (int*)carve((size_t)N_EDGES * sizeof(int));
    int*   partials = (int*)carve(512 * sizeof(int));
    float* gbuf     = (float*)carve((size_t)N_GRAPHS * DOUTF * sizeof(float));
    float* bufA     = (float*)carve((size_t)N_NODES * DH * sizeof(float));
    float* bufB     = (float*)carve((size_t)N_NODES * DH * sizeof(float));

    const int* src = ei;
    const int* dst = ei + N_EDGES;

    // ---- build CSR (dst -> list of src), per call (deterministic)
    k_zero<<<256, 256, 0, stream>>>(rowptr, N_NODES + 1);
    k_zero<<<256, 256, 0, stream>>>(cursor, N_NODES);
    k_zero<<<256, 256, 0, stream>>>((int*)gbuf, N_GRAPHS * DOUTF);

    int eblocks = (N_EDGES + 255) / 256;
    k_hist<<<eblocks, 256, 0, stream>>>(dst, rowptr, N_EDGES);
    int sblocks = (N_NODES + 1 + 255) / 256;
    k_scan1<<<sblocks, 256, 0, stream>>>(rowptr, N_NODES + 1, partials);
    k_scan2<<<1, 1, 0, stream>>>(partials, sblocks);
    k_scan3<<<sblocks, 256, 0, stream>>>(rowptr, N_NODES + 1, partials);
    k_fill<<<eblocks, 256, 0, stream>>>(src, dst, rowptr, cursor, colbuf, N_EDGES);

    // ---- 5 fused GIN layers (ping-pong activation buffers)
    int gblocks = (N_NODES + 63) / 64;
    k_gin<DIN0><<<gblocks, 256, 0, stream>>>(x,    gw1[0], gb1[0], gw2[0], gb2[0], rowptr, colbuf, bufA, N_NODES);
    k_gin<DH>  <<<gblocks, 256, 0, stream>>>(bufA, gw1[1], gb1[1], gw2[1], gb2[1], rowptr, colbuf, bufB, N_NODES);
    k_gin<DH>  <<<gblocks, 256, 0, stream>>>(bufB, gw1[2], gb1[2], gw2[2], gb2[2], rowptr, colbuf, bufA, N_NODES);
    k_gin<DH>  <<<gblocks, 256, 0, stream>>>(bufA, gw1[3], gb1[3], gw2[3], gb2[3], rowptr, colbuf, bufB, N_NODES);
    k_gin<DH>  <<<gblocks, 256, 0, stream>>>(bufB, gw1[4], gb1[4], gw2[4], gb2[4], rowptr, colbuf, bufA, N_NODES);

    // ---- readout + layernorm
    int rblocks = (N_NODES + 255) / 256;
    k_readout<<<rblocks, 128, 0, stream>>>(bufA, lw, lb, batch, gbuf, N_NODES);
    k_ln<<<N_GRAPHS, 128, 0, stream>>>(gbuf, lng, lnb, out);

    (void)in_sizes; (void)n_in; (void)out_size; (void)ws_size;
}